// PatchAttention_7971459301803
// MI455X (gfx1250) — compile-verified
//
#include <hip/hip_runtime.h>
#include <math.h>

#define D_MODEL 256
#define NHEAD 8
#define HEAD_DIM 32
#define NTOK 512
#define QKV_COLS (3 * D_MODEL) /* 768 */

typedef float v2f __attribute__((ext_vector_type(2)));
typedef float v8f __attribute__((ext_vector_type(8)));

__device__ __forceinline__ float silu_f(float x) {
    return x * (1.0f / (1.0f + __expf(-x)));
}

// ---------------------------------------------------------------------------
// Stage 1: qkv = x @ Wqkv^T + bqkv   (M=512, N=768, K=256)
// One wave per 16x16 tile, V_WMMA_F32_16X16X4_F32, K stepped by 4.
// A[m][k] = x[m0+m][k0+k]; B[k][n] = Wqkv[n0+n][k0+k]  (Wqkv is (768,256) RM)
// ---------------------------------------------------------------------------
__global__ void qkv_gemm_kernel(const float* __restrict__ x,
                                const float* __restrict__ Wqkv,
                                const float* __restrict__ bqkv,
                                float* __restrict__ qkv) {
    const int lane = threadIdx.x;     // 0..31
    const int r16  = lane & 15;
    const int kh   = lane >> 4;       // 0/1 -> K half
    const int m0 = blockIdx.x * 16;
    const int n0 = blockIdx.y * 16;

    const float* arow = x    + (size_t)(m0 + r16) * D_MODEL;
    const float* brow = Wqkv + (size_t)(n0 + r16) * D_MODEL;

    v8f acc = {};
    #pragma unroll 8
    for (int k0 = 0; k0 < D_MODEL; k0 += 4) {
        v2f a = *(const v2f*)(arow + k0 + 2 * kh);   // (m=lane%16, k=2*kh+r)
        v2f b = *(const v2f*)(brow + k0 + 2 * kh);   // (k=2*kh+r, n=lane%16)
        acc = __builtin_amdgcn_wmma_f32_16x16x4_f32(
            /*neg_a=*/false, a, /*neg_b=*/false, b,
            /*c_mod=*/(short)0, acc, /*reuse_a=*/false, /*reuse_b=*/false);
    }

    const int col = n0 + r16;
    const float bias = bqkv[col];
    #pragma unroll
    for (int r = 0; r < 8; ++r) {
        const int row = m0 + r + 8 * kh;
        qkv[(size_t)row * QKV_COLS + col] = acc[r] + bias;
    }
}

// ---------------------------------------------------------------------------
// Stage 2: scores[h][i][j] = (q_i . k_j) / sqrt(32)   (per head K=32)
// One wave per (16i x 16j) tile per head.
// ---------------------------------------------------------------------------
__global__ void scores_gemm_kernel(const float* __restrict__ qkv,
                                   float* __restrict__ scores) {
    const int lane = threadIdx.x;
    const int r16  = lane & 15;
    const int kh   = lane >> 4;
    const int i0 = blockIdx.x * 16;
    const int j0 = blockIdx.y * 16;
    const int h  = blockIdx.z;

    const float* qrow = qkv + (size_t)(i0 + r16) * QKV_COLS + h * HEAD_DIM;            // Q
    const float* krow = qkv + (size_t)(j0 + r16) * QKV_COLS + D_MODEL + h * HEAD_DIM;  // K

    v8f acc = {};
    #pragma unroll
    for (int k0 = 0; k0 < HEAD_DIM; k0 += 4) {
        v2f a = *(const v2f*)(qrow + k0 + 2 * kh);
        v2f b = *(const v2f*)(krow + k0 + 2 * kh);  // B[k][j] = K[j][k] -> Q.K^T
        acc = __builtin_amdgcn_wmma_f32_16x16x4_f32(
            false, a, false, b, (short)0, acc, false, false);
    }

    const float scale = 0.17677669529663687f; // 1/sqrt(32)
    float* shead = scores + (size_t)h * NTOK * NTOK;
    #pragma unroll
    for (int r = 0; r < 8; ++r) {
        const int row = i0 + r + 8 * kh;
        shead[(size_t)row * NTOK + (j0 + r16)] = acc[r] * scale;
    }
}

// ---------------------------------------------------------------------------
// Stage 3: scores[h][i][j] += (silu(rel_ij @ Wb1^T + bb1) @ Wb2^T + bb2)[h]
// One thread per (i,j) pair; the 256-wide hidden loop is VALU FMA + v_exp.
// Weight loads are wave-uniform -> scalar path.
// ---------------------------------------------------------------------------
__global__ void relbias_kernel(const float* __restrict__ coords,
                               const float* __restrict__ Wb1,
                               const float* __restrict__ bb1,
                               const float* __restrict__ Wb2,
                               const float* __restrict__ bb2,
                               float* __restrict__ scores) {
    const int p = blockIdx.x * blockDim.x + threadIdx.x;
    const int i = p >> 9;
    const int j = p & (NTOK - 1);
    const float dx = coords[i * 3 + 0] - coords[j * 3 + 0];
    const float dy = coords[i * 3 + 1] - coords[j * 3 + 1];
    const float dz = coords[i * 3 + 2] - coords[j * 3 + 2];

    float bacc[NHEAD];
    #pragma unroll
    for (int h = 0; h < NHEAD; ++h) bacc[h] = bb2[h];

    for (int c = 0; c < D_MODEL; ++c) {
        const float hb = silu_f(fmaf(Wb1[c * 3 + 0], dx,
                              fmaf(Wb1[c * 3 + 1], dy,
                              fmaf(Wb1[c * 3 + 2], dz, bb1[c]))));
        #pragma unroll
        for (int h = 0; h < NHEAD; ++h)
            bacc[h] = fmaf(hb, Wb2[h * D_MODEL + c], bacc[h]);
    }
    #pragma unroll
    for (int h = 0; h < NHEAD; ++h)
        scores[((size_t)h * NTOK + i) * NTOK + j] += bacc[h];
}

// ---------------------------------------------------------------------------
// Stage 4: in-place softmax over j for each (h,i) row. One wave per row.
// ---------------------------------------------------------------------------
__global__ void softmax_kernel(float* __restrict__ scores) {
    float* s = scores + (size_t)blockIdx.x * NTOK;
    const int lane = threadIdx.x;

    float vals[16];
    float m = -INFINITY;
    #pragma unroll
    for (int t = 0; t < 16; ++t) {
        vals[t] = s[lane + t * 32];
        m = fmaxf(m, vals[t]);
    }
    #pragma unroll
    for (int off = 16; off > 0; off >>= 1)
        m = fmaxf(m, __shfl_xor(m, off, 32));

    float sum = 0.0f;
    #pragma unroll
    for (int t = 0; t < 16; ++t) {
        vals[t] = __expf(vals[t] - m);
        sum += vals[t];
    }
    #pragma unroll
    for (int off = 16; off > 0; off >>= 1)
        sum += __shfl_xor(sum, off, 32);

    const float inv = 1.0f / sum;
    #pragma unroll
    for (int t = 0; t < 16; ++t)
        s[lane + t * 32] = vals[t] * inv;
}

// ---------------------------------------------------------------------------
// Stage 5: fused context. Uses sum_j w == 1 identity:
//   out[i, h*32+d] = sum_j w[h,i,j]*v[h,j,d]
//                  + (sum_j w[h,i,j]*hv[i,j,:]) @ Wv2[h*32+d,:] + bv2[h*32+d]
// One block (256 threads) per query i; thread owns hidden channel c
// (and output channel c = h*32+d).
// ---------------------------------------------------------------------------
__global__ void context_kernel(const float* __restrict__ qkv,
                               const float* __restrict__ weights, // (H,N,N) post-softmax
                               const float* __restrict__ coords,
                               const float* __restrict__ Wv1,
                               const float* __restrict__ bv1,
                               const float* __restrict__ Wv2,
                               const float* __restrict__ bv2,
                               float* __restrict__ out) {
    __shared__ float wrow[NHEAD * NTOK];    // 16 KB: all heads' weight rows for i
    __shared__ float gbuf[NHEAD * D_MODEL]; //  8 KB: g[h][c]

    const int i = blockIdx.x;
    const int c = threadIdx.x;   // 0..255
    const int h = c >> 5;        // this thread's output head

    for (int idx = c; idx < NHEAD * NTOK; idx += D_MODEL) {
        const int hh = idx >> 9;
        const int jj = idx & (NTOK - 1);
        wrow[idx] = weights[((size_t)hh * NTOK + i) * NTOK + jj];
    }
    __syncthreads();

    const float w0 = Wv1[c * 3 + 0], w1 = Wv1[c * 3 + 1], w2 = Wv1[c * 3 + 2];
    const float bc = bv1[c];
    const float cix = coords[i * 3 + 0], ciy = coords[i * 3 + 1], ciz = coords[i * 3 + 2];

    float g[NHEAD] = {};
    float base = 0.0f;
    for (int j = 0; j < NTOK; ++j) {
        const float dx = cix - coords[j * 3 + 0];
        const float dy = ciy - coords[j * 3 + 1];
        const float dz = ciz - coords[j * 3 + 2];
        const float hv = silu_f(fmaf(w0, dx, fmaf(w1, dy, fmaf(w2, dz, bc))));
        #pragma unroll
        for (int hh = 0; hh < NHEAD; ++hh)
            g[hh] = fmaf(wrow[hh * NTOK + j], hv, g[hh]);
        base = fmaf(wrow[h * NTOK + j],
                    qkv[(size_t)j * QKV_COLS + 2 * D_MODEL + c], base);
    }

    #pragma unroll
    for (int hh = 0; hh < NHEAD; ++hh)
        gbuf[hh * D_MODEL + c] = g[hh];
    __syncthreads();

    // rel_context projection: out channel c corresponds to Wv2 row c (= h*32+d)
    float rc = bv2[c];
    const float* w2row = Wv2 + (size_t)c * D_MODEL;
    const float* ghead = gbuf + h * D_MODEL;
    for (int cc = 0; cc < D_MODEL; ++cc)
        rc = fmaf(ghead[cc], w2row[cc], rc);

    out[(size_t)i * D_MODEL + c] = base + rc;
}

// ---------------------------------------------------------------------------
extern "C" void kernel_launch(void* const* d_in, const int* in_sizes, int n_in,
                              void* d_out, int out_size, void* d_ws, size_t ws_size,
                              hipStream_t stream) {
    const float* x      = (const float*)d_in[0];
    const float* coords = (const float*)d_in[1];
    const float* Wqkv   = (const float*)d_in[2];
    const float* bqkv   = (const float*)d_in[3];
    const float* Wb1    = (const float*)d_in[4];
    const float* bb1    = (const float*)d_in[5];
    const float* Wb2    = (const float*)d_in[6];
    const float* bb2    = (const float*)d_in[7];
    const float* Wv1    = (const float*)d_in[8];
    const float* bv1    = (const float*)d_in[9];
    const float* Wv2    = (const float*)d_in[10];
    const float* bv2    = (const float*)d_in[11];
    float* out = (float*)d_out;

    char* ws = (char*)d_ws;
    float* qkv    = (float*)ws;                                              // 1.5 MB
    float* scores = (float*)(ws + (size_t)NTOK * QKV_COLS * sizeof(float));  // 8 MB

    qkv_gemm_kernel<<<dim3(NTOK / 16, QKV_COLS / 16), 32, 0, stream>>>(x, Wqkv, bqkv, qkv);
    scores_gemm_kernel<<<dim3(NTOK / 16, NTOK / 16, NHEAD), 32, 0, stream>>>(qkv, scores);
    relbias_kernel<<<dim3((NTOK * NTOK) / 256), 256, 0, stream>>>(coords, Wb1, bb1, Wb2, bb2, scores);
    softmax_kernel<<<dim3(NHEAD * NTOK), 32, 0, stream>>>(scores);
    context_kernel<<<dim3(NTOK), D_MODEL, 0, stream>>>(qkv, scores, coords, Wv1, bv1, Wv2, bv2, out);
}